// WindowAttention_47090021433596
// MI455X (gfx1250) — compile-verified
//
#include <hip/hip_runtime.h>
#include <hip/hip_bf16.h>
#include <math.h>

typedef float v2f __attribute__((ext_vector_type(2)));
typedef float v8f __attribute__((ext_vector_type(8)));

#define NTOK 64
#define NHEADS 6
#define HD 32
#define DIM 192
#define ROWSTRIDE (3 * DIM) /* 576 floats per token row of qkv */
#define LSTR 36 /* LDS row stride (floats) for q/k/km: 36m mod 64 hits all multiples of 4 -> conflict-free b64 frags */
#define VSTR 40 /* LDS row stride for v/vm: 2*40 mod 64 = 16 -> half-wave row groups use disjoint bank ranges */
#define ASTR 66 /* LDS row stride for P scratch: even (8B-aligned b64 loads), 66m mod 64 = 2m distinct */

// ---------------- bias MLP: [64,64,2] -> relu(@W1+b1) -> @W2+b2 -> [6,64,64] ----------------
__global__ void bias_mlp_kernel(const float* __restrict__ W1, const float* __restrict__ b1,
                                const float* __restrict__ W2, const float* __restrict__ b2,
                                float* __restrict__ bias_out) {
  int idx = blockIdx.x * blockDim.x + threadIdx.x;
  if (idx >= NTOK * NTOK) return;
  int n = idx >> 6, m = idx & 63;
  float d0 = (float)((n >> 3) - (m >> 3));
  float d1 = (float)((n & 7) - (m & 7));
  float r0 = (d0 > 0.f ? 1.f : (d0 < 0.f ? -1.f : 0.f)) * log1pf(fabsf(d0));
  float r1 = (d1 > 0.f ? 1.f : (d1 < 0.f ? -1.f : 0.f)) * log1pf(fabsf(d1));
  float acc[NHEADS];
#pragma unroll
  for (int h = 0; h < NHEADS; ++h) acc[h] = 0.f;
  for (int k = 0; k < 256; ++k) {
    float hv = fmaf(r0, W1[k], fmaf(r1, W1[256 + k], b1[k]));
    hv = fmaxf(hv, 0.f);
#pragma unroll
    for (int h = 0; h < NHEADS; ++h) acc[h] = fmaf(hv, W2[k * NHEADS + h], acc[h]);
  }
#pragma unroll
  for (int h = 0; h < NHEADS; ++h) bias_out[h * NTOK * NTOK + idx] = acc[h] + b2[h];
}

// ---------------- fused dual-stream window attention, f32 WMMA ----------------
__launch_bounds__(128)
__global__ void win_attn_kernel(const float* __restrict__ qkv,
                                const float* __restrict__ qkv_mfif,
                                const float* __restrict__ bias,
                                const int* __restrict__ tag_p,
                                float* __restrict__ out, int nwin) {
  __shared__ float s_q[NTOK * LSTR];
  __shared__ float s_k[NTOK * LSTR];
  __shared__ float s_km[NTOK * LSTR];
  __shared__ float s_v[NTOK * VSTR];
  __shared__ float s_vm[NTOK * VSTR];
  __shared__ float s_attn[NTOK * ASTR];

  const int b = blockIdx.x / NHEADS;
  const int h = blockIdx.x % NHEADS;
  const int tid = threadIdx.x;
  const int lane = tid & 31;
  const int wave = tid >> 5;
  const int tag = *tag_p;
  const float scale = 0.17677669529663687f; // 32^-0.5

  // tag==2: both streams use qm (from qkv_mfif); tag==1: both use q (from qkv)
  const float* qsrc = (tag == 2) ? qkv_mfif : qkv;
  const size_t wbase = (size_t)b * NTOK * ROWSTRIDE;
  const float* gq  = qsrc     + wbase + (0 * NHEADS + h) * HD;
  const float* gk  = qkv      + wbase + (1 * NHEADS + h) * HD;
  const float* gkm = qkv_mfif + wbase + (1 * NHEADS + h) * HD;
  const float* gv  = qkv      + wbase + (2 * NHEADS + h) * HD;
  const float* gvm = qkv_mfif + wbase + (2 * NHEADS + h) * HD;

  // stage 5 matrices of 64x32 f32 (512 float4 each, 4 per thread)
#pragma unroll
  for (int i = 0; i < 4; ++i) {
    int j = tid + i * 128;
    int row = j >> 3;
    int c4 = (j & 7) << 2;
    const int go = row * ROWSTRIDE + c4;
    float4 fq = *(const float4*)(gq + go);
    fq.x *= scale; fq.y *= scale; fq.z *= scale; fq.w *= scale;
    *(float4*)(&s_q [row * LSTR + c4]) = fq;
    *(float4*)(&s_k [row * LSTR + c4]) = *(const float4*)(gk + go);
    *(float4*)(&s_km[row * LSTR + c4]) = *(const float4*)(gkm + go);
    *(float4*)(&s_v [row * VSTR + c4]) = *(const float4*)(gv + go);
    *(float4*)(&s_vm[row * VSTR + c4]) = *(const float4*)(gvm + go);
  }
  __syncthreads();

  const int mrow = lane & 15;            // M (or N) coordinate within a 16x16 tile
  const int khalf = (lane >> 4) << 1;    // K sub-offset: lanes 0-15 -> 0, 16-31 -> 2
  const int arow = wave * 16 + mrow;     // this wave's stripe row for A fragments
  const int rowoff = wave * 16 + ((lane >> 4) << 3); // C/D layout row base (VGPR r -> row rowoff+r)

  // preload bias fragments for this wave's C-tile coordinates (shared by both streams)
  float bfr[4][8];
  {
    const float* bh = bias + h * NTOK * NTOK;
#pragma unroll
    for (int t = 0; t < 4; ++t)
#pragma unroll
      for (int r = 0; r < 8; ++r)
        bfr[t][r] = bh[(rowoff + r) * NTOK + t * 16 + mrow];
  }

  const v8f zero8 = {0.f, 0.f, 0.f, 0.f, 0.f, 0.f, 0.f, 0.f};

#pragma unroll
  for (int stream = 0; stream < 2; ++stream) {
    const float* s_kk = stream ? s_km : s_k;
    const float* s_vv = stream ? s_vm : s_v;

    // ---- S = q·K^T : wave computes 16x64 stripe as 4 tiles, K=32 ----
    v8f acc[4] = {zero8, zero8, zero8, zero8};
#pragma unroll
    for (int kk = 0; kk < HD; kk += 4) {
      v2f a = *(const v2f*)(&s_q[arow * LSTR + kk + khalf]);
#pragma unroll
      for (int t = 0; t < 4; ++t) {
        v2f bfrag = *(const v2f*)(&s_kk[(t * 16 + mrow) * LSTR + kk + khalf]);
        acc[t] = __builtin_amdgcn_wmma_f32_16x16x4_f32(false, a, false, bfrag,
                                                       (short)0, acc[t], false, false);
      }
    }

    // ---- bias add + in-register softmax (each row lives across one 16-lane half) ----
#pragma unroll
    for (int r = 0; r < 8; ++r) {
      float v0 = acc[0][r] + bfr[0][r];
      float v1 = acc[1][r] + bfr[1][r];
      float v2 = acc[2][r] + bfr[2][r];
      float v3 = acc[3][r] + bfr[3][r];
      float mx = fmaxf(fmaxf(v0, v1), fmaxf(v2, v3));
#pragma unroll
      for (int off = 1; off < 16; off <<= 1)
        mx = fmaxf(mx, __shfl_xor(mx, off, 32));
      v0 = __expf(v0 - mx); v1 = __expf(v1 - mx);
      v2 = __expf(v2 - mx); v3 = __expf(v3 - mx);
      float sum = v0 + v1 + v2 + v3;
#pragma unroll
      for (int off = 1; off < 16; off <<= 1)
        sum += __shfl_xor(sum, off, 32);
      float inv = 1.0f / sum;
      acc[0][r] = v0 * inv; acc[1][r] = v1 * inv;
      acc[2][r] = v2 * inv; acc[3][r] = v3 * inv;
    }

    // ---- spill P to wave-private LDS rows (C-layout -> A-layout conversion) ----
#pragma unroll
    for (int r = 0; r < 8; ++r)
#pragma unroll
      for (int t = 0; t < 4; ++t)
        s_attn[(rowoff + r) * ASTR + t * 16 + mrow] = acc[t][r];

    // ---- X = P·V : 16x64 @ 64x32, 2 output tiles, K=64 ----
    v8f xacc[2] = {zero8, zero8};
#pragma unroll
    for (int kk = 0; kk < NTOK; kk += 4) {
      v2f a = *(const v2f*)(&s_attn[arow * ASTR + kk + khalf]);
#pragma unroll
      for (int t = 0; t < 2; ++t) {
        v2f bfrag;
        bfrag.x = s_vv[(kk + khalf) * VSTR + t * 16 + mrow];
        bfrag.y = s_vv[(kk + khalf + 1) * VSTR + t * 16 + mrow];
        xacc[t] = __builtin_amdgcn_wmma_f32_16x16x4_f32(false, a, false, bfrag,
                                                        (short)0, xacc[t], false, false);
      }
    }

    // ---- write stripe: out[b, n, h*32 + d]; stream 0 = x_ivif, 1 = x_mfif ----
    float* op = out + (size_t)stream * ((size_t)nwin * NTOK * DIM)
                    + (size_t)b * NTOK * DIM + h * HD;
#pragma unroll
    for (int r = 0; r < 8; ++r)
#pragma unroll
      for (int t = 0; t < 2; ++t)
        op[(rowoff + r) * DIM + t * 16 + mrow] = xacc[t][r];

    __syncthreads(); // s_attn reuse hygiene between streams
  }
}

extern "C" void kernel_launch(void* const* d_in, const int* in_sizes, int n_in,
                              void* d_out, int out_size, void* d_ws, size_t ws_size,
                              hipStream_t stream) {
  const float* qkv      = (const float*)d_in[0];
  const float* qkv_mfif = (const float*)d_in[1];
  const float* W1 = (const float*)d_in[2];
  const float* b1 = (const float*)d_in[3];
  const float* W2 = (const float*)d_in[4];
  const float* b2 = (const float*)d_in[5];
  const int* tag  = (const int*)d_in[6];
  float* out = (float*)d_out;
  float* biasbuf = (float*)d_ws; // 6*64*64 f32 = 98304 B

  int nwin = in_sizes[0] / (NTOK * ROWSTRIDE); // 2048

  bias_mlp_kernel<<<(NTOK * NTOK + 255) / 256, 256, 0, stream>>>(W1, b1, W2, b2, biasbuf);
  win_attn_kernel<<<nwin * NHEADS, 128, 0, stream>>>(qkv, qkv_mfif, biasbuf, tag, out, nwin);
}